// STGI_65094524338517
// MI455X (gfx1250) — compile-verified
//
#include <hip/hip_runtime.h>
#include <math.h>

typedef __attribute__((ext_vector_type(2))) float v2f;
typedef __attribute__((ext_vector_type(8))) float v8f;

#define T_DIM   12
#define N_NODES 10000
#define F_DIM   32
#define H_DIM   64
#define E_EDGES 320000
#define TN      (T_DIM * N_NODES)          // 120000, divisible by 16

// ---------------------------------------------------------------- zero fill
__global__ void stgi_zero_kernel(float* __restrict__ p, long long n) {
    long long i = (long long)blockIdx.x * blockDim.x + threadIdx.x;
    long long stride = (long long)gridDim.x * blockDim.x;
    for (; i < n; i += stride) p[i] = 0.0f;
}

// ---------------------------------------------------------------- degree
__global__ void stgi_deg_kernel(const long long* __restrict__ col,
                                const float* __restrict__ ew,
                                float* __restrict__ deg, int E) {
    int e = blockIdx.x * blockDim.x + threadIdx.x;
    if (e < E) atomicAdd(&deg[col[e]], ew[e]);
}

// deg -> dinv, in place (matches reference: deg>0 ? rsqrt(max(deg,1e-12)) : 0)
__global__ void stgi_dinv_kernel(float* __restrict__ deg, int n) {
    int i = blockIdx.x * blockDim.x + threadIdx.x;
    if (i < n) {
        float d = deg[i];
        deg[i] = (d > 0.0f) ? rsqrtf(fmaxf(d, 1e-12f)) : 0.0f;
    }
}

__global__ void stgi_norm_kernel(const long long* __restrict__ row,
                                 const long long* __restrict__ col,
                                 const float* __restrict__ ew,
                                 const float* __restrict__ dinv,
                                 float* __restrict__ norm, int E) {
    int e = blockIdx.x * blockDim.x + threadIdx.x;
    if (e < E) norm[e] = dinv[row[e]] * ew[e] * dinv[col[e]];
}

// ------------------------------------------------- edge aggregation (F=32 lanes)
// One wave32 per edge; lane = feature (F_DIM == 32). Loops over T inside so the
// edge indices/norm are loaded once. Gather at row, scatter-atomic-add at col.
__global__ void stgi_aggregate_kernel(const float* __restrict__ src,   // (T,N,F)
                                      const long long* __restrict__ row,
                                      const long long* __restrict__ col,
                                      const float* __restrict__ norm,
                                      float* __restrict__ dst,         // (T,N,F) zeroed
                                      int E) {
    int w    = (blockIdx.x * blockDim.x + threadIdx.x) >> 5;   // edge id
    int lane = threadIdx.x & 31;                               // feature id
    if (w >= E) return;
    long long r = row[w];
    long long c = col[w];
    float nm = norm[w];
    const float* s = src + r * F_DIM + lane;
    float*       d = dst + c * F_DIM + lane;
    const long long step = (long long)N_NODES * F_DIM;
#pragma unroll
    for (int t = 0; t < T_DIM; ++t) {
        float v = s[(long long)t * step] * nm;
        atomicAdd(&d[(long long)t * step], v);
    }
}

// ------------------------------------------------- WMMA fp32 GEMM
// out(M,Nout) = A(M,K) @ W(K,Nout) [+ bias] [relu], one wave per 16x16 tile,
// V_WMMA_F32_16X16X4_F32 with K stepped by 4.
// A-frag layout (ISA 7.12.2, 32-bit A 16x4):   lane<16: K=k0,k0+1 ; lane>=16: K=k0+2,k0+3 ; M=lane&15
// B-frag layout (symmetric, 4x16):             lane<16: K=k0,k0+1 ; lane>=16: K=k0+2,k0+3 ; N=lane&15
// C/D layout: VGPR v -> M = v + 8*(lane>=16), N = lane&15
__global__ void stgi_wmma_gemm_kernel(const float* __restrict__ A,
                                      const float* __restrict__ W,
                                      const float* __restrict__ bias,   // may be null
                                      float* __restrict__ out,
                                      int M, int K, int Nout, int do_relu) {
    int wave = (blockIdx.x * blockDim.x + threadIdx.x) >> 5;
    int lane = threadIdx.x & 31;
    int ntiles = Nout >> 4;
    int tm = wave / ntiles;
    int tn = wave - tm * ntiles;
    if (tm * 16 >= M) return;              // wave-uniform; EXEC stays all-1s

    int half = lane >> 4;                  // 0 or 1
    int ml   = lane & 15;

    const float* Arow = A + (long long)(tm * 16 + ml) * K;   // this lane's A row
    const float* Bcol = W + tn * 16 + ml;                    // this lane's B column

    v8f c = {};
    for (int k0 = 0; k0 < K; k0 += 4) {
        v2f a, b;
        int ka = k0 + 2 * half;
        a.x = Arow[ka];
        a.y = Arow[ka + 1];
        b.x = Bcol[(long long)ka * Nout];
        b.y = Bcol[(long long)(ka + 1) * Nout];
        c = __builtin_amdgcn_wmma_f32_16x16x4_f32(
                /*neg_a=*/false, a, /*neg_b=*/false, b,
                /*c_mod=*/(short)0, c, /*reuse_a=*/false, /*reuse_b=*/false);
    }

    float bv = bias ? bias[tn * 16 + ml] : 0.0f;
    float* obase = out + (long long)(tm * 16 + 8 * half) * Nout + tn * 16 + ml;
#pragma unroll
    for (int v = 0; v < 8; ++v) {
        float x = c[v] + bv;
        if (do_relu) x = fmaxf(x, 0.0f);
        obase[(long long)v * Nout] = x;
    }
}

// ------------------------------------------------- final bias + tanh
__global__ void stgi_finish_kernel(float* __restrict__ out,
                                   const float* __restrict__ b2, long long n) {
    long long i = (long long)blockIdx.x * blockDim.x + threadIdx.x;
    if (i < n) out[i] = tanhf(out[i] + b2[i & (F_DIM - 1)]);
}

// ----------------------------------------------------------------- driver
extern "C" void kernel_launch(void* const* d_in, const int* in_sizes, int n_in,
                              void* d_out, int out_size, void* d_ws, size_t ws_size,
                              hipStream_t stream) {
    const float*     x   = (const float*)d_in[0];       // (T,N,F)
    const long long* ei  = (const long long*)d_in[1];   // (2,E) int64
    const float*     ew  = (const float*)d_in[2];       // (E,)
    // d_in[3] missing_mask: unused by reference math
    const float*     W1  = (const float*)d_in[4];       // (F,H)
    const float*     b1  = (const float*)d_in[5];       // (H,)
    const float*     W2  = (const float*)d_in[6];       // (H,F)
    const float*     b2  = (const float*)d_in[7];       // (F,)
    float*           out = (float*)d_out;               // (T,N,F)

    const long long* row = ei;            // sources
    const long long* col = ei + E_EDGES;  // destinations

    // workspace layout (floats): deg/dinv | norm | agg1 (reused as h2) | out1
    float* ws   = (float*)d_ws;
    float* deg  = ws;                                   // N
    float* norm = deg + N_NODES;                        // E
    float* agg1 = norm + E_EDGES;                       // T*N*F  (later reused as h2)
    float* out1 = agg1 + (long long)TN * F_DIM;         // T*N*H
    (void)ws_size; (void)in_sizes; (void)n_in; (void)out_size;

    const int BLK = 256;

    // 1) degree -> dinv -> per-edge norm
    stgi_zero_kernel<<<(N_NODES + BLK - 1) / BLK, BLK, 0, stream>>>(deg, N_NODES);
    stgi_deg_kernel<<<(E_EDGES + BLK - 1) / BLK, BLK, 0, stream>>>(col, ew, deg, E_EDGES);
    stgi_dinv_kernel<<<(N_NODES + BLK - 1) / BLK, BLK, 0, stream>>>(deg, N_NODES);
    stgi_norm_kernel<<<(E_EDGES + BLK - 1) / BLK, BLK, 0, stream>>>(row, col, ew, deg, norm, E_EDGES);

    // 2) layer 1, aggregate-first (A x): messages 32-wide instead of 64-wide
    stgi_zero_kernel<<<2048, BLK, 0, stream>>>(agg1, (long long)TN * F_DIM);
    stgi_aggregate_kernel<<<(E_EDGES * 32) / BLK, BLK, 0, stream>>>(x, row, col, norm, agg1, E_EDGES);

    // 3) out1 = relu((A x) W1 + b1)   — WMMA f32, M=120000 K=32 Nout=64
    {
        int waves = (TN / 16) * (H_DIM / 16);           // 30000
        stgi_wmma_gemm_kernel<<<(waves * 32 + BLK - 1) / BLK, BLK, 0, stream>>>(
            agg1, W1, b1, out1, TN, F_DIM, H_DIM, /*relu=*/1);
    }

    // 4) h2 = out1 @ W2 (narrow 64->32 BEFORE edge pass) — reuse agg1 buffer
    float* h2 = agg1;
    {
        int waves = (TN / 16) * (F_DIM / 16);           // 15000
        stgi_wmma_gemm_kernel<<<(waves * 32 + BLK - 1) / BLK, BLK, 0, stream>>>(
            out1, W2, nullptr, h2, TN, H_DIM, F_DIM, /*relu=*/0);
    }

    // 5) layer-2 aggregation into d_out, then tanh(out + b2)
    stgi_zero_kernel<<<2048, BLK, 0, stream>>>(out, (long long)TN * F_DIM);
    stgi_aggregate_kernel<<<(E_EDGES * 32) / BLK, BLK, 0, stream>>>(h2, row, col, norm, out, E_EDGES);
    stgi_finish_kernel<<<((long long)TN * F_DIM + BLK - 1) / BLK, BLK, 0, stream>>>(
        out, b2, (long long)TN * F_DIM);
}